// IISSM_83897891160150
// MI455X (gfx1250) — compile-verified
//
#include <hip/hip_runtime.h>
#include <cmath>

#define B_  2
#define C_  96
#define H_  48
#define W_  48
#define L_  (H_*W_)          // 2304
#define HEADS_ 8
#define DK_ 12
#define BASEC_ 32
#define DSTATE_ 16
#define DCONV_ 4
#define DINNER_ 192
#define DTRANK_ 6
#define BD_ 8                // 4 directions * B
#define XPROJ_N 38           // dt_rank + 2*d_state
#define XPROJ_P 48           // padded to multiple of 16
#define GEMM_WAVES 4

typedef __attribute__((ext_vector_type(2))) float v2f;
typedef __attribute__((ext_vector_type(8))) float v8f;

static __device__ __forceinline__ v8f wmma_f32(v2f a, v2f b, v8f c) {
  // D = A(16x4 f32) * B(4x16 f32) + C(16x16 f32); exact fp32 matrix pipe.
  return __builtin_amdgcn_wmma_f32_16x16x4_f32(false, a, false, b, (short)0, c, false, false);
}
static __device__ __forceinline__ float silu_f(float x){ return x / (1.f + __expf(-x)); }
static __device__ __forceinline__ float softplus_f(float x){ return x > 20.f ? x : log1pf(__expf(x)); }
static __device__ __forceinline__ v2f ldv2(const float* p){ return *reinterpret_cast<const v2f*>(p); }

// ---------------------------------------------------------------------------
// Guard-free WMMA fp32 GEMM: Out[m,n] = sum_k A[m,k]*W[n,k] (+bias[n]) (+resid)
// REQUIRES: M % (16*GEMM_WAVES) == 0, N % 16 == 0, K % 4 == 0, A/W 8B-aligned
// with even leading strides. Out row-major [M,ldo] unless store_bcl!=0, in
// which case row m = b*Lsp + l and Out[(b*Cch+n)*Lsp + l] (+ residual).
// ---------------------------------------------------------------------------
__global__ void gemm_wmma_kernel(const float* __restrict__ A, int lda,
                                 const float* __restrict__ W, int ldw,
                                 float* __restrict__ Out, int ldo,
                                 int M, int N, int K,
                                 const float* __restrict__ bias,
                                 const float* __restrict__ resid_bcl,
                                 int Lsp, int Cch, int store_bcl)
{
  const int lane = threadIdx.x & 31;
  const int wave = threadIdx.x >> 5;
  const int tileN = blockIdx.x;
  const int tileM = blockIdx.y * GEMM_WAVES + wave;
  const int half = lane >> 4;
  const int lan  = lane & 15;
  const int m = tileM*16 + lan;
  const int n = tileN*16 + lan;
  const float* Ap = A + (size_t)m*lda + half*2;
  const float* Wp = W + (size_t)n*ldw + half*2;
  v8f acc = {0.f,0.f,0.f,0.f,0.f,0.f,0.f,0.f};
  for (int k0 = 0; k0 < K; k0 += 4) {
    const v2f av = ldv2(Ap + k0);
    const v2f bv = ldv2(Wp + k0);
    acc = wmma_f32(av, bv, acc);
  }
  const int nn = tileN*16 + lan;
  const float badd = bias ? bias[nn] : 0.f;
  if (store_bcl) {
#pragma unroll
    for (int r = 0; r < 8; ++r) {
      const int mm = tileM*16 + r + half*8;
      const int bb = mm / Lsp, ll = mm % Lsp;
      const size_t oidx = ((size_t)bb*Cch + nn)*(size_t)Lsp + ll;
      Out[oidx] = acc[r] + badd + (resid_bcl ? resid_bcl[oidx] : 0.f);
    }
  } else {
#pragma unroll
    for (int r = 0; r < 8; ++r) {
      const int mm = tileM*16 + r + half*8;
      Out[(size_t)mm*ldo + nn] = acc[r] + badd;
    }
  }
}

// ---------------------------------------------------------------------------
// Zero-pad m_xproj_w [38,192] -> [48,192]
// ---------------------------------------------------------------------------
__global__ void pad_xproj_kernel(const float* __restrict__ w, float* __restrict__ wp)
{
  const int idx = blockIdx.x*blockDim.x + threadIdx.x;
  if (idx >= XPROJ_P*DINNER_) return;
  const int nrow = idx / DINNER_;
  wp[idx] = (nrow < XPROJ_N) ? w[idx] : 0.f;
}

// ---------------------------------------------------------------------------
// LayerNorm over C=96 at each (b,l); x is [B,C,L], out xn is [B*L, 96].
// ---------------------------------------------------------------------------
__global__ void ln_kernel(const float* __restrict__ x, const float* __restrict__ w,
                          const float* __restrict__ bias, float* __restrict__ xn)
{
  const int idx = blockIdx.x*blockDim.x + threadIdx.x;
  if (idx >= B_*L_) return;
  const int b = idx / L_, l = idx % L_;
  const float* xp = x + (size_t)b*C_*L_ + l;
  float mu = 0.f;
  for (int c = 0; c < C_; ++c) mu += xp[(size_t)c*L_];
  mu *= (1.f/C_);
  float var = 0.f;
  for (int c = 0; c < C_; ++c) { float d = xp[(size_t)c*L_] - mu; var += d*d; }
  var *= (1.f/C_);
  const float inv = rsqrtf(var + 1e-5f);
  for (int c = 0; c < C_; ++c)
    xn[(size_t)idx*C_ + c] = (xp[(size_t)c*L_] - mu)*inv*w[c] + bias[c];
}

// ---------------------------------------------------------------------------
// Bilinear resize f_lu [B,32,12,12] -> fr [B*L, 32] (half-pixel centers, x4)
// ---------------------------------------------------------------------------
__global__ void resize_kernel(const float* __restrict__ f, float* __restrict__ fr)
{
  const int idx = blockIdx.x*blockDim.x + threadIdx.x;
  if (idx >= B_*L_*BASEC_) return;
  const int c  = idx & 31;
  const int rl = idx >> 5;
  const int b  = rl / L_, l = rl % L_;
  const int yy = l / W_, xx = l % W_;
  float sy = (yy + 0.5f)*0.25f - 0.5f;
  float sx = (xx + 0.5f)*0.25f - 0.5f;
  sy = fminf(fmaxf(sy, 0.f), 11.f);
  sx = fminf(fmaxf(sx, 0.f), 11.f);
  int y0 = (int)sy, x0 = (int)sx;
  const float wy = sy - y0, wx = sx - x0;
  const int y1 = min(y0+1, 11), x1 = min(x0+1, 11);
  const float* fp = f + ((size_t)b*BASEC_ + c)*144;
  const float v00 = fp[y0*12+x0], v01 = fp[y0*12+x1];
  const float v10 = fp[y1*12+x0], v11 = fp[y1*12+x1];
  fr[(size_t)rl*BASEC_ + c] = (v00*(1.f-wx)+v01*wx)*(1.f-wy) + (v10*(1.f-wx)+v11*wx)*wy;
}

// ---------------------------------------------------------------------------
// Flash attention: one wave per (b, h, 16-query tile). q/k/v: [B*L, 96].
// ---------------------------------------------------------------------------
__global__ void attn_kernel(const float* __restrict__ q, const float* __restrict__ k,
                            const float* __restrict__ v, float* __restrict__ o)
{
  __shared__ float sP[8][16][17];
  const int wave = threadIdx.x >> 5;
  const int lane = threadIdx.x & 31;
  const int qt = blockIdx.x * 8 + wave;     // 0..143
  const int h  = blockIdx.y;
  const int b  = blockIdx.z;
  const int half = lane >> 4, lan = lane & 15;
  const float scale = 0.28867513459481287f; // 1/sqrt(12)
  const float vmask = (lan < DK_) ? 1.f : 0.f;
  const int   vch   = h*DK_ + min(lan, DK_-1);   // clamped channel, no exec branch

  v2f qa[3];
  const float* qp = q + ((size_t)b*L_ + qt*16 + lan)*C_ + h*DK_ + half*2;
#pragma unroll
  for (int s = 0; s < 3; ++s) qa[s] = ldv2(qp + s*4);   // d0 max = 11 < 12

  float mrow[8], lrow[8];
#pragma unroll
  for (int r = 0; r < 8; ++r) { mrow[r] = -1e30f; lrow[r] = 0.f; }
  v8f Oacc = {0.f,0.f,0.f,0.f,0.f,0.f,0.f,0.f};

  for (int kt = 0; kt < L_/16; ++kt) {
    // S tile = Q K^T via fp32 WMMA (K = 12 -> 3 steps)
    v8f s8 = {0.f,0.f,0.f,0.f,0.f,0.f,0.f,0.f};
    const float* kp = k + ((size_t)b*L_ + kt*16 + lan)*C_ + h*DK_ + half*2;
#pragma unroll
    for (int s = 0; s < 3; ++s)
      s8 = wmma_f32(qa[s], ldv2(kp + s*4), s8);

    // online softmax; row m = r + 8*half lives in the 16 lanes of this half
#pragma unroll
    for (int r = 0; r < 8; ++r) {
      float sc = s8[r] * scale;
      float rm = sc;
      rm = fmaxf(rm, __shfl_xor(rm, 1, 32));
      rm = fmaxf(rm, __shfl_xor(rm, 2, 32));
      rm = fmaxf(rm, __shfl_xor(rm, 4, 32));
      rm = fmaxf(rm, __shfl_xor(rm, 8, 32));
      const float mnew = fmaxf(mrow[r], rm);
      const float p    = __expf(sc - mnew);
      const float corr = __expf(mrow[r] - mnew);
      float ps = p;
      ps += __shfl_xor(ps, 1, 32);
      ps += __shfl_xor(ps, 2, 32);
      ps += __shfl_xor(ps, 4, 32);
      ps += __shfl_xor(ps, 8, 32);
      lrow[r] = lrow[r]*corr + ps;
      mrow[r] = mnew;
      Oacc[r] *= corr;                       // same row mapping as S
      sP[wave][r + half*8][lan] = p;         // LDS round-trip to A-layout
    }
    asm volatile("s_wait_dscnt 0" ::: "memory");
    // O += P V  (K = 16 keys -> 4 WMMA steps), C operand = running Oacc
    const size_t vb = (size_t)b*L_ + kt*16;
#pragma unroll
    for (int s = 0; s < 4; ++s) {
      const int n0 = s*4 + half*2;
      v2f av; av.x = sP[wave][lan][n0]; av.y = sP[wave][lan][n0+1];
      v2f bv;
      bv.x = v[(vb + n0    )*C_ + vch] * vmask;   // unconditional load + cndmask
      bv.y = v[(vb + n0 + 1)*C_ + vch] * vmask;
      Oacc = wmma_f32(av, bv, Oacc);
    }
  }
#pragma unroll
  for (int r = 0; r < 8; ++r) {
    const int m = r + half*8;
    if (lan < DK_)
      o[((size_t)b*L_ + qt*16 + m)*C_ + h*DK_ + lan] = Oacc[r] / lrow[r];
  }
}

// ---------------------------------------------------------------------------
// Build 4-direction sequence z[8, L, 96] from o2 [B*L, 96]
// ---------------------------------------------------------------------------
__global__ void zbuild_kernel(const float* __restrict__ o2, float* __restrict__ z)
{
  const int idx = blockIdx.x*blockDim.x + threadIdx.x;
  if (idx >= BD_*L_*C_) return;
  const int c  = idx % C_;
  const int t  = idx / C_;
  const int l  = t % L_;
  const int bd = t / L_;
  const int dir = bd / B_, b = bd % B_;
  int y = l / W_, x = l % W_;
  if (dir & 1) x = W_-1-x;
  if (dir & 2) y = H_-1-y;
  z[idx] = o2[((size_t)b*L_ + y*W_ + x)*C_ + c];
}

// ---------------------------------------------------------------------------
// Depthwise causal conv (4 taps) + SiLU: xz[.,:192] -> xc [BD*L, 192]
// ---------------------------------------------------------------------------
__global__ void conv_kernel(const float* __restrict__ xz, const float* __restrict__ cw,
                            const float* __restrict__ cb, float* __restrict__ xc)
{
  const int idx = blockIdx.x*blockDim.x + threadIdx.x;
  if (idx >= BD_*L_*DINNER_) return;
  const int d  = idx % DINNER_;
  const int t  = idx / DINNER_;
  const int l  = t % L_;
  const int bd = t / L_;
  float sum = cb[d];
#pragma unroll
  for (int j = 0; j < DCONV_; ++j) {
    const int ls = l - (DCONV_-1) + j;
    if (ls >= 0) sum += xz[((size_t)bd*L_ + ls)*(2*DINNER_) + d] * cw[d*DCONV_ + j];
  }
  xc[(size_t)t*DINNER_ + d] = silu_f(sum);
}

// ---------------------------------------------------------------------------
// dt = softplus(x_dbl[:, :6] @ dtw^T + dtb)   (K=6, scalar path)
// ---------------------------------------------------------------------------
__global__ void dt_kernel(const float* __restrict__ xdbl, const float* __restrict__ dtw,
                          const float* __restrict__ dtb, float* __restrict__ dt)
{
  const int idx = blockIdx.x*blockDim.x + threadIdx.x;
  if (idx >= BD_*L_*DINNER_) return;
  const int d   = idx % DINNER_;
  const int row = idx / DINNER_;
  float acc = dtb[d];
#pragma unroll
  for (int r = 0; r < DTRANK_; ++r)
    acc += xdbl[(size_t)row*XPROJ_P + r] * dtw[d*DTRANK_ + r];
  dt[idx] = softplus_f(acc);
}

// ---------------------------------------------------------------------------
// Selective scan: lane = (d&1, n); h = e^{dt*A} h + dt*B*x; y = <h,C> + D*x
// 768 waves total (8 bd * 96 d-pairs), 2304 sequential steps each.
// ---------------------------------------------------------------------------
__global__ void scan_kernel(const float* __restrict__ dt, const float* __restrict__ xc,
                            const float* __restrict__ xdbl, const float* __restrict__ Alog,
                            const float* __restrict__ Dp, float* __restrict__ y)
{
  const int wave = threadIdx.x >> 5;
  const int lane = threadIdx.x & 31;
  const int wg = blockIdx.x * 8 + wave;      // 0..767
  const int bd = wg / 96;
  const int dp = wg % 96;
  const int half = lane >> 4;
  const int n = lane & 15;
  const int d = dp*2 + half;
  const float Aval = -__expf(Alog[d*DSTATE_ + n]);
  const float Dd = Dp[d];
  float h = 0.f;
  for (int t = 0; t < L_; ++t) {
    const size_t row = (size_t)bd*L_ + t;
    const float dtv = dt[row*DINNER_ + d];
    const float xcv = xc[row*DINNER_ + d];
    const float Bn  = xdbl[row*XPROJ_P + DTRANK_ + n];
    const float Cn  = xdbl[row*XPROJ_P + DTRANK_ + DSTATE_ + n];
    h = __expf(dtv*Aval)*h + dtv*Bn*xcv;
    float s = h*Cn;
    s += __shfl_xor(s, 1, 32);
    s += __shfl_xor(s, 2, 32);
    s += __shfl_xor(s, 4, 32);
    s += __shfl_xor(s, 8, 32);
    if (n == 0) y[row*DINNER_ + d] = s + Dd*xcv;
  }
}

// ---------------------------------------------------------------------------
// yg = y * silu(zg), zg = xz[:, 192:384]
// ---------------------------------------------------------------------------
__global__ void gate_kernel(const float* __restrict__ y, const float* __restrict__ xz,
                            float* __restrict__ yg)
{
  const int idx = blockIdx.x*blockDim.x + threadIdx.x;
  if (idx >= BD_*L_*DINNER_) return;
  const int d   = idx % DINNER_;
  const int row = idx / DINNER_;
  yg[idx] = y[idx] * silu_f(xz[(size_t)row*(2*DINNER_) + DINNER_ + d]);
}

// ---------------------------------------------------------------------------
// Combine 4 directions (inverse flips) -> yc [B*L, 96]
// ---------------------------------------------------------------------------
__global__ void combine_kernel(const float* __restrict__ ym, float* __restrict__ yc)
{
  const int idx = blockIdx.x*blockDim.x + threadIdx.x;
  if (idx >= B_*L_*C_) return;
  const int c = idx % C_;
  const int t = idx / C_;
  const int b = t / L_, l = t % L_;
  const int yy = l / W_, xx = l % W_;
  const int l1 = yy*W_ + (W_-1-xx);
  const int l2 = (H_-1-yy)*W_ + xx;
  const int l3 = (H_-1-yy)*W_ + (W_-1-xx);
  const float s =
      ym[(((size_t)0*B_ + b)*L_ + l )*C_ + c] +
      ym[(((size_t)1*B_ + b)*L_ + l1)*C_ + c] +
      ym[(((size_t)2*B_ + b)*L_ + l2)*C_ + c] +
      ym[(((size_t)3*B_ + b)*L_ + l3)*C_ + c];
  yc[idx] = 0.25f * s;
}

// ---------------------------------------------------------------------------
static inline int cdiv(int a, int b) { return (a + b - 1) / b; }

extern "C" void kernel_launch(void* const* d_in, const int* in_sizes, int n_in,
                              void* d_out, int out_size, void* d_ws, size_t ws_size,
                              hipStream_t stream)
{
  (void)in_sizes; (void)n_in; (void)out_size; (void)ws_size;
  const float* x       = (const float*)d_in[0];
  const float* f_lu    = (const float*)d_in[1];
  const float* norm_w  = (const float*)d_in[2];
  const float* norm_b  = (const float*)d_in[3];
  const float* qproj_w = (const float*)d_in[4];
  const float* qproj_b = (const float*)d_in[5];
  const float* iq_w    = (const float*)d_in[6];
  const float* iq_b    = (const float*)d_in[7];
  const float* ik_w    = (const float*)d_in[8];
  const float* ik_b    = (const float*)d_in[9];
  const float* iv_w    = (const float*)d_in[10];
  const float* iv_b    = (const float*)d_in[11];
  const float* io_w    = (const float*)d_in[12];
  const float* io_b    = (const float*)d_in[13];
  const float* m_in_w  = (const float*)d_in[14];
  const float* m_conv_w= (const float*)d_in[15];
  const float* m_conv_b= (const float*)d_in[16];
  const float* m_xproj = (const float*)d_in[17];
  const float* m_dtw   = (const float*)d_in[18];
  const float* m_dtb   = (const float*)d_in[19];
  const float* m_Alog  = (const float*)d_in[20];
  const float* m_D     = (const float*)d_in[21];
  const float* m_out_w = (const float*)d_in[22];
  const float* out_w   = (const float*)d_in[23];
  const float* out_b   = (const float*)d_in[24];
  float* out = (float*)d_out;

  // workspace layout (floats)
  float* ws = (float*)d_ws;
  size_t off = 0;
  float* xn   = ws + off; off += (size_t)B_*L_*C_;          // 442368
  float* fr   = ws + off; off += (size_t)B_*L_*BASEC_;      // 147456
  float* fq   = ws + off; off += (size_t)B_*L_*C_;
  float* qb   = ws + off; off += (size_t)B_*L_*C_;
  float* kb   = ws + off; off += (size_t)B_*L_*C_;
  float* vb   = ws + off; off += (size_t)B_*L_*C_;
  float* ob   = ws + off; off += (size_t)B_*L_*C_;
  float* o2   = ws + off; off += (size_t)B_*L_*C_;
  float* zb   = ws + off; off += (size_t)BD_*L_*C_;         // 1769472
  float* xz   = ws + off; off += (size_t)BD_*L_*2*DINNER_;  // 7077888
  float* xc   = ws + off; off += (size_t)BD_*L_*DINNER_;    // 3538944
  float* xdbl = ws + off; off += (size_t)BD_*L_*XPROJ_P;    // 884736
  float* xprojp = ws + off; off += (size_t)XPROJ_P*DINNER_; // 9216 (padded weights)
  float* dtb  = ws + off; off += (size_t)BD_*L_*DINNER_;    // reused as yg after scan
  float* yb   = ws + off; off += (size_t)BD_*L_*DINNER_;
  float* ym   = ws + off; off += (size_t)BD_*L_*C_;
  float* yc   = ws + off; off += (size_t)B_*L_*C_;          // total ~25.8M floats

  const int M1 = B_*L_;      // 4608
  const int M2 = BD_*L_;     // 18432
  const dim3 gblk(128);

  // 0. pad x_proj weights to N=48
  pad_xproj_kernel<<<cdiv(XPROJ_P*DINNER_,256), 256, 0, stream>>>(m_xproj, xprojp);
  // 1. LayerNorm
  ln_kernel<<<cdiv(M1,256), 256, 0, stream>>>(x, norm_w, norm_b, xn);
  // 2. bilinear resize
  resize_kernel<<<cdiv(M1*BASEC_,256), 256, 0, stream>>>(f_lu, fr);
  // 3. fq = fr @ qproj^T + b
  gemm_wmma_kernel<<<dim3(C_/16, M1/(16*GEMM_WAVES)), gblk, 0, stream>>>(
      fr, BASEC_, qproj_w, BASEC_, fq, C_, M1, C_, BASEC_, qproj_b, nullptr, L_, C_, 0);
  // 4-6. q, k, v projections
  gemm_wmma_kernel<<<dim3(C_/16, M1/(16*GEMM_WAVES)), gblk, 0, stream>>>(
      fq, C_, iq_w, C_, qb, C_, M1, C_, C_, iq_b, nullptr, L_, C_, 0);
  gemm_wmma_kernel<<<dim3(C_/16, M1/(16*GEMM_WAVES)), gblk, 0, stream>>>(
      xn, C_, ik_w, C_, kb, C_, M1, C_, C_, ik_b, nullptr, L_, C_, 0);
  gemm_wmma_kernel<<<dim3(C_/16, M1/(16*GEMM_WAVES)), gblk, 0, stream>>>(
      xn, C_, iv_w, C_, vb, C_, M1, C_, C_, iv_b, nullptr, L_, C_, 0);
  // 7. flash attention (wave per 16-query tile)
  attn_kernel<<<dim3(L_/16/8, HEADS_, B_), 256, 0, stream>>>(qb, kb, vb, ob);
  // 8. io projection
  gemm_wmma_kernel<<<dim3(C_/16, M1/(16*GEMM_WAVES)), gblk, 0, stream>>>(
      ob, C_, io_w, C_, o2, C_, M1, C_, C_, io_b, nullptr, L_, C_, 0);
  // 9. 4-direction gather
  zbuild_kernel<<<cdiv(BD_*L_*C_,256), 256, 0, stream>>>(o2, zb);
  // 10. xz = z @ m_in_w^T  (N=384)
  gemm_wmma_kernel<<<dim3((2*DINNER_)/16, M2/(16*GEMM_WAVES)), gblk, 0, stream>>>(
      zb, C_, m_in_w, C_, xz, 2*DINNER_, M2, 2*DINNER_, C_, nullptr, nullptr, L_, C_, 0);
  // 11. depthwise conv + silu
  conv_kernel<<<cdiv(M2*DINNER_,256), 256, 0, stream>>>(xz, m_conv_w, m_conv_b, xc);
  // 12. x_dbl = xc @ m_xproj_pad^T (N=48, cols 38..47 are zero)
  gemm_wmma_kernel<<<dim3(XPROJ_P/16, M2/(16*GEMM_WAVES)), gblk, 0, stream>>>(
      xc, DINNER_, xprojp, DINNER_, xdbl, XPROJ_P, M2, XPROJ_P, DINNER_, nullptr, nullptr, L_, C_, 0);
  // 13. dt = softplus(dtproj)
  dt_kernel<<<cdiv(M2*DINNER_,256), 256, 0, stream>>>(xdbl, m_dtw, m_dtb, dtb);
  // 14. selective scan
  scan_kernel<<<96, 256, 0, stream>>>(dtb, xc, xdbl, m_Alog, m_D, yb);
  // 15. gate (reuse dt buffer as yg)
  float* yg = dtb;
  gate_kernel<<<cdiv(M2*DINNER_,256), 256, 0, stream>>>(yb, xz, yg);
  // 16. ym = yg @ m_out_w^T
  gemm_wmma_kernel<<<dim3(C_/16, M2/(16*GEMM_WAVES)), gblk, 0, stream>>>(
      yg, DINNER_, m_out_w, DINNER_, ym, C_, M2, C_, DINNER_, nullptr, nullptr, L_, C_, 0);
  // 17. combine directions
  combine_kernel<<<cdiv(B_*L_*C_,256), 256, 0, stream>>>(ym, yc);
  // 18. final projection + bias + residual, stored in [B,C,H,W]
  gemm_wmma_kernel<<<dim3(C_/16, M1/(16*GEMM_WAVES)), gblk, 0, stream>>>(
      yc, C_, out_w, C_, out, C_, M1, C_, C_, out_b, x, L_, C_, 1);
}